// AttnDecoder_55937654063436
// MI455X (gfx1250) — compile-verified
//
#include <hip/hip_runtime.h>
#include <math.h>

typedef __attribute__((ext_vector_type(2))) float v2f;
typedef __attribute__((ext_vector_type(8))) float v8f;

#define HDIM 1024
#define EDIM 1024
#define LCNN 20
#define VOCAB 50257

__device__ __forceinline__ float wave_reduce_add(float v) {
#pragma unroll
  for (int m = 16; m > 0; m >>= 1) v += __shfl_xor(v, m, 32);
  return v;
}

// ---------------------------------------------------------------------------
// GEMV via V_WMMA_F32_16X16X4_F32.
// out[r] = dot(W[r,:], x) (+ bias[r]) (+ emb[y*HDIM + r])
// One wave per 16 rows. A tile: lane l holds row (rows0 + (l&15)),
// K = (l>>4)*2 + {0,1}  -> one 8B load per lane per k-step (each weight row
// streamed exactly once). B tile: x-chunk replicated across all 16 N columns
// (per-lane value depends only on K = (l>>4)*2 + v). After the K sweep,
// lane 0 holds rows 0..7 and lane 16 rows 8..15 (N=0 column of C/D layout).
//
// Cache policy: REGULAR temporal (no NT). Total weight footprint ~250MB vs
// 192MB global L2 -> most weights stay L2-resident across timed graph
// replays; NT would discard that. Unroll 8 for deeper load pipelining on the
// cold pass (more outstanding global_load_b64 per wave).
// ---------------------------------------------------------------------------
__global__ void gemv16_wmma(const float* __restrict__ W, const float* __restrict__ x,
                            const float* __restrict__ bias,
                            const float* __restrict__ emb, const int* __restrict__ yptr,
                            float* __restrict__ out, int R, int K) {
  const int lane  = threadIdx.x & 31;
  const int wave  = (blockIdx.x * blockDim.x + threadIdx.x) >> 5;
  const int rows0 = wave * 16;
  if (rows0 >= R) return;                      // wave-uniform: EXEC stays all-ones
  const int m    = lane & 15;
  const int half = lane >> 4;
  const int row  = min(rows0 + m, R - 1);      // clamp so OOB-tile loads stay valid
  const float* Wr = W + (size_t)row * K + half * 2;
  const float* xr = x + half * 2;

  v8f acc = {0.f, 0.f, 0.f, 0.f, 0.f, 0.f, 0.f, 0.f};
#pragma unroll 8
  for (int kb = 0; kb < K; kb += 4) {
    v2f a = *(const v2f*)(Wr + kb);            // weights: keep L2-resident
    v2f b = *(const v2f*)(xr + kb);            // vector: hot in cache
    acc = __builtin_amdgcn_wmma_f32_16x16x4_f32(
        /*neg_a=*/false, a, /*neg_b=*/false, b,
        /*c_mod=*/(short)0, acc, /*reuse_a=*/false, /*reuse_b=*/false);
  }

  if (m == 0) {                                // lanes 0 and 16 own the results
    const int rb = rows0 + half * 8;
#pragma unroll
    for (int i = 0; i < 8; ++i) {
      const int r = rb + i;
      if (r < R) {
        float v = acc[i];
        if (bias) v += bias[r];
        if (emb)  v += emb[(size_t)yptr[0] * HDIM + r];
        out[r] = v;
      }
    }
  }
}

// ---------------------------------------------------------------------------
// Attention: scores[l] = sum_e d[e]*cnn_a[e,l]; softmax over L=20;
// c[e] = sum_l a[l]*cnn_c[e,l].  Also builds x = [emb[y], c] for the GRU.
// Single block (tiny: ~160KB traffic).
// ---------------------------------------------------------------------------
__global__ void attn_ctx(const float* __restrict__ dvec, const float* __restrict__ cnn_a,
                         const float* __restrict__ cnn_c, const float* __restrict__ emb,
                         const int* __restrict__ yptr, float* __restrict__ xbuf) {
  __shared__ float s_sc[LCNN];
  __shared__ float s_a[LCNN];
  const int tid = threadIdx.x;
  if (tid < LCNN) s_sc[tid] = 0.f;
  __syncthreads();

  float ps[LCNN];
#pragma unroll
  for (int l = 0; l < LCNN; ++l) ps[l] = 0.f;
  for (int e = tid; e < EDIM; e += blockDim.x) {
    const float de = dvec[e];
    const float* ar = cnn_a + e * LCNN;
#pragma unroll
    for (int l = 0; l < LCNN; ++l) ps[l] += de * ar[l];
  }
#pragma unroll
  for (int l = 0; l < LCNN; ++l) {
    float v = wave_reduce_add(ps[l]);
    if ((tid & 31) == 0) atomicAdd(&s_sc[l], v);   // ds_add_f32
  }
  __syncthreads();

  if (tid == 0) {
    float mx = s_sc[0];
    for (int l = 1; l < LCNN; ++l) mx = fmaxf(mx, s_sc[l]);
    float sum = 0.f;
    for (int l = 0; l < LCNN; ++l) { float ex = expf(s_sc[l] - mx); s_a[l] = ex; sum += ex; }
    const float inv = 1.f / sum;
    for (int l = 0; l < LCNN; ++l) s_a[l] *= inv;
  }
  __syncthreads();

  const size_t yoff = (size_t)yptr[0] * HDIM;
  for (int e = tid; e < EDIM; e += blockDim.x) {
    const float* cr = cnn_c + e * LCNN;
    float ce = 0.f;
#pragma unroll
    for (int l = 0; l < LCNN; ++l) ce += s_a[l] * cr[l];
    xbuf[HDIM + e] = ce;             // context half of x
    xbuf[e]        = emb[yoff + e];  // embedding half of x
  }
}

// GRU gate combine (PyTorch layout r,z,n): h_new = (1-z)*n + z*h
__global__ void gru_combine(const float* __restrict__ gi, const float* __restrict__ gh,
                            const float* __restrict__ h, float* __restrict__ hnew,
                            float* __restrict__ outh) {
  const int i = blockIdx.x * blockDim.x + threadIdx.x;
  if (i >= HDIM) return;
  const float r = 1.f / (1.f + expf(-(gi[i] + gh[i])));
  const float z = 1.f / (1.f + expf(-(gi[HDIM + i] + gh[HDIM + i])));
  const float n = tanhf(gi[2 * HDIM + i] + r * gh[2 * HDIM + i]);
  const float hv = (1.f - z) * n + z * h[i];
  hnew[i] = hv;
  outh[i] = hv;
}

// In-place log-softmax over V elements; one block of 1024 threads (32 waves).
__global__ void logsoftmax_inplace(float* __restrict__ p, int n) {
  __shared__ float red[32];
  __shared__ float s_val;
  const int tid = threadIdx.x;
  const int nw  = blockDim.x >> 5;

  float mx = -INFINITY;
  for (int i = tid; i < n; i += blockDim.x) mx = fmaxf(mx, p[i]);
#pragma unroll
  for (int m = 16; m > 0; m >>= 1) mx = fmaxf(mx, __shfl_xor(mx, m, 32));
  if ((tid & 31) == 0) red[tid >> 5] = mx;
  __syncthreads();
  if (tid == 0) {
    float v = red[0];
    for (int w = 1; w < nw; ++w) v = fmaxf(v, red[w]);
    s_val = v;
  }
  __syncthreads();
  const float gmax = s_val;

  float sum = 0.f;
  for (int i = tid; i < n; i += blockDim.x) sum += expf(p[i] - gmax);
  sum = wave_reduce_add(sum);
  if ((tid & 31) == 0) red[tid >> 5] = sum;
  __syncthreads();
  if (tid == 0) {
    float v = 0.f;
    for (int w = 0; w < nw; ++w) v += red[w];
    s_val = gmax + logf(v);
  }
  __syncthreads();
  const float lse = s_val;
  for (int i = tid; i < n; i += blockDim.x) p[i] -= lse;
}

extern "C" void kernel_launch(void* const* d_in, const int* in_sizes, int n_in,
                              void* d_out, int out_size, void* d_ws, size_t ws_size,
                              hipStream_t stream) {
  const int*   y     = (const int*)  d_in[0];
  const float* h     = (const float*)d_in[1];
  const float* cnn_a = (const float*)d_in[2];
  const float* cnn_c = (const float*)d_in[3];
  const float* emb   = (const float*)d_in[4];
  const float* W_t   = (const float*)d_in[5];
  const float* b_t   = (const float*)d_in[6];
  const float* W_ih  = (const float*)d_in[7];
  const float* W_hh  = (const float*)d_in[8];
  const float* b_ih  = (const float*)d_in[9];
  const float* b_hh  = (const float*)d_in[10];
  const float* W_o   = (const float*)d_in[11];
  const float* b_o   = (const float*)d_in[12];
  float* out = (float*)d_out;

  float* ws   = (float*)d_ws;
  float* dvec = ws;             // 1024
  float* xbuf = ws + 1024;      // 2048  ([g, c])
  float* gi   = ws + 3072;      // 3072
  float* gh   = ws + 6144;      // 3072
  float* hnew = ws + 9216;      // 1024

  auto blocks_for = [](int R) { int tiles = (R + 15) / 16; return (tiles + 7) / 8; };

  // d = W_t @ h + b_t + emb[y]
  gemv16_wmma<<<blocks_for(EDIM), 256, 0, stream>>>(W_t, h, b_t, emb, y, dvec, EDIM, HDIM);
  // attention -> x = [g, c]
  attn_ctx<<<1, 256, 0, stream>>>(dvec, cnn_a, cnn_c, emb, y, xbuf);
  // gi = W_ih @ x + b_ih ; gh = W_hh @ h + b_hh
  gemv16_wmma<<<blocks_for(3 * HDIM), 256, 0, stream>>>(W_ih, xbuf, b_ih, nullptr, nullptr, gi, 3 * HDIM, 2 * HDIM);
  gemv16_wmma<<<blocks_for(3 * HDIM), 256, 0, stream>>>(W_hh, h,    b_hh, nullptr, nullptr, gh, 3 * HDIM, HDIM);
  // h_new
  gru_combine<<<(HDIM + 255) / 256, 256, 0, stream>>>(gi, gh, h, hnew, out + VOCAB);
  // logits = W_o @ h_new + b_o  (written straight into d_out)
  gemv16_wmma<<<blocks_for(VOCAB), 256, 0, stream>>>(W_o, hnew, b_o, nullptr, nullptr, out, VOCAB, HDIM);
  // log-softmax in place
  logsoftmax_inplace<<<1, 1024, 0, stream>>>(out, VOCAB);
}